// slotGATConv_1700807049270
// MI455X (gfx1250) — compile-verified
//
#include <hip/hip_runtime.h>

#define N_NODES 50000
#define E_EDGES 800000
#define NTYPE   3
#define IN_F    64
#define OUT_F   16
#define NHEAD   4
#define EDGE_F  64
#define NETYPE  5
#define SLOPE   0.2f
#define ROW_F   (NTYPE * IN_F)           // 192: feat row stride
#define FS_F    (NHEAD * NTYPE * OUT_F)  // 192: fs / rst row stride
#define SEG_F   (NTYPE * OUT_F)          // 48: per-head feature count
#define NH      (N_NODES * NHEAD)
#define EH      (E_EDGES * NHEAD)

typedef __attribute__((ext_vector_type(2))) float v2f;
typedef __attribute__((ext_vector_type(8))) float v8f;

#if __has_builtin(__builtin_amdgcn_wmma_f32_16x16x4_f32)
#define USE_WMMA_F32X4 1
#else
#define USE_WMMA_F32X4 0
#endif

// Monotonic float <-> uint key so atomicMax(uint) implements float max.
__device__ __forceinline__ unsigned int fkey(float x) {
  unsigned int b = __float_as_uint(x);
  return (b & 0x80000000u) ? ~b : (b | 0x80000000u);
}
__device__ __forceinline__ float funkey(unsigned int k) {
  unsigned int b = (k & 0x80000000u) ? (k & 0x7FFFFFFFu) : ~k;
  return __uint_as_float(b);
}

// ---------------------------------------------------------------- init
__global__ void k_init(float* __restrict__ rst, float* __restrict__ ssum,
                       unsigned int* __restrict__ mkey) {
  int tid = blockIdx.x * blockDim.x + threadIdx.x;
  if (tid < N_NODES * FS_F) rst[tid] = 0.0f;
  if (tid < NH) {
    ssum[tid] = 0.0f;
    mkey[tid] = 0x007FFFFFu;  // fkey(-inf)
  }
}

// ------------------------------------------- per-edge-type attention bias
// ee_type[et][h] = sum_f attn_e[h][f] * sum_k edge_emb[et][k] * fc_e_w[h*64+f][k]
__global__ void k_ee(const float* __restrict__ edge_emb,
                     const float* __restrict__ fc_e_w,
                     const float* __restrict__ attn_e,
                     float* __restrict__ ee_type) {
  int tid = threadIdx.x;
  if (tid >= NETYPE * NHEAD) return;
  int et = tid / NHEAD, h = tid % NHEAD;
  const float* emb = edge_emb + (size_t)et * EDGE_F;
  float acc = 0.0f;
  for (int f = 0; f < EDGE_F; ++f) {
    const float* wrow = fc_e_w + (size_t)(h * EDGE_F + f) * EDGE_F;
    float dot = 0.0f;
    for (int k = 0; k < EDGE_F; ++k) dot += emb[k] * wrow[k];
    acc += attn_e[h * EDGE_F + f] * dot;
  }
  ee_type[tid] = acc;  // layout [et][h] flat = et*4+h
}

// ---------------------------------------------------- slot projection (WMMA)
// fs[n][h][t*16+o] = sum_i feat[n][t*64+i] * fc[t][i][h*16+o]
// One wave per (16-node tile, slot). 16 K-steps x 4 head tiles of f32 WMMA.
__global__ __launch_bounds__(128) void k_proj(const float* __restrict__ feat,
                                              const float* __restrict__ fc,
                                              float* __restrict__ fs) {
  const int lane = threadIdx.x & 31;
  const int wid = blockIdx.x * (blockDim.x >> 5) + (threadIdx.x >> 5);
  const int NWAVES = (N_NODES / 16) * NTYPE;  // 50000 % 16 == 0
  if (wid >= NWAVES) return;                  // wave-uniform: EXEC stays all-1s
  const int t = wid % NTYPE;
  const int node0 = (wid / NTYPE) * 16;
  const int half = lane >> 4;   // K-half selector per ISA A/B layout
  const int m = lane & 15;      // M for A-frag, N for B/C-frag

#if USE_WMMA_F32X4
  // A fragments: lane holds rows node0+m, K = 4k + 2*half + {0,1}
  const float* arow = feat + (size_t)(node0 + m) * ROW_F + t * IN_F + half * 2;
  v2f afrag[16];
#pragma unroll
  for (int k = 0; k < 16; ++k) {
    afrag[k].x = arow[k * 4 + 0];
    afrag[k].y = arow[k * 4 + 1];
  }
  const float* bbase = fc + (size_t)t * IN_F * (NHEAD * OUT_F);
#pragma unroll
  for (int j = 0; j < NHEAD; ++j) {  // each 16-col N-tile == one head
    v8f acc = {};
#pragma unroll
    for (int k = 0; k < 16; ++k) {
      const float* bp = bbase + (size_t)(k * 4 + half * 2) * (NHEAD * OUT_F)
                        + j * OUT_F + m;
      v2f bfrag;
      bfrag.x = bp[0];
      bfrag.y = bp[NHEAD * OUT_F];
      acc = __builtin_amdgcn_wmma_f32_16x16x4_f32(
          false, afrag[k], false, bfrag, (short)0, acc, false, false);
    }
    // C layout: VGPR r, lane -> row M = r + 8*half, col N = m
    float* out = fs + (size_t)node0 * FS_F + j * SEG_F + t * OUT_F + m;
#pragma unroll
    for (int r = 0; r < 8; ++r) {
      out[(size_t)(r + 8 * half) * FS_F] = acc[r];
    }
  }
#else
  // scalar fallback (keeps file compiling if the f32x4 builtin is absent)
  const float* bbase = fc + (size_t)t * IN_F * (NHEAD * OUT_F);
  for (int j = 0; j < NHEAD; ++j) {
    for (int r = 0; r < 8; ++r) {
      int M = r + 8 * half;
      const float* ap = feat + (size_t)(node0 + M) * ROW_F + t * IN_F;
      const float* bp = bbase + j * OUT_F + m;
      float s = 0.0f;
      for (int k = 0; k < IN_F; ++k) s += ap[k] * bp[(size_t)k * (NHEAD * OUT_F)];
      fs[(size_t)(node0 + M) * FS_F + j * SEG_F + t * OUT_F + m] = s;
    }
  }
#endif
}

// ------------------------------------------------------ el / er node dots
__global__ void k_elr(const float* __restrict__ fs,
                      const float* __restrict__ attn_l,
                      const float* __restrict__ attn_r,
                      float* __restrict__ el, float* __restrict__ er) {
  int tid = blockIdx.x * blockDim.x + threadIdx.x;  // tid = n*4 + h
  if (tid >= NH) return;
  int h = tid & 3;
  const float4* row = (const float4*)(fs + (size_t)(tid >> 2) * FS_F + h * SEG_F);
  const float* al = attn_l + h * SEG_F;
  const float* ar = attn_r + h * SEG_F;
  float sl = 0.0f, sr = 0.0f;
#pragma unroll
  for (int c = 0; c < SEG_F / 4; ++c) {
    float4 v = row[c];
    sl += v.x * al[4 * c + 0] + v.y * al[4 * c + 1] + v.z * al[4 * c + 2] + v.w * al[4 * c + 3];
    sr += v.x * ar[4 * c + 0] + v.y * ar[4 * c + 1] + v.z * ar[4 * c + 2] + v.w * ar[4 * c + 3];
  }
  el[tid] = sl;
  er[tid] = sr;
}

// --------------------------------------- edge scores + leaky relu + seg max
__global__ void k_score(const int* __restrict__ src, const int* __restrict__ dst,
                        const int* __restrict__ etype,
                        const float* __restrict__ el, const float* __restrict__ er,
                        const float* __restrict__ ee_type,
                        float* __restrict__ aout, unsigned int* __restrict__ mkey) {
  int tid = blockIdx.x * blockDim.x + threadIdx.x;  // tid = e*4 + h
  if (tid >= EH) return;
  int e = tid >> 2, h = tid & 3;
  int s = src[e], d = dst[e], t = etype[e];
  float sc = el[s * NHEAD + h] + er[d * NHEAD + h] + ee_type[t * NHEAD + h];
  sc = sc > 0.0f ? sc : SLOPE * sc;
  aout[tid] = sc;
  atomicMax(&mkey[d * NHEAD + h], fkey(sc));
}

// ----------------------------------------------------- exp + segment sum
__global__ void k_expsum(const int* __restrict__ dst, float* __restrict__ aout,
                         const unsigned int* __restrict__ mkey,
                         float* __restrict__ ssum) {
  int tid = blockIdx.x * blockDim.x + threadIdx.x;
  if (tid >= EH) return;
  int e = tid >> 2, h = tid & 3;
  int d = dst[e];
  float ex = __expf(aout[tid] - funkey(mkey[d * NHEAD + h]));
  aout[tid] = ex;
  atomicAdd(&ssum[d * NHEAD + h], ex);
}

// -------------------------------- normalize + gather-scale-scatter messages
__global__ void k_msg(const int* __restrict__ src, const int* __restrict__ dst,
                      float* __restrict__ aout, const float* __restrict__ ssum,
                      const float* __restrict__ fs, float* __restrict__ rst) {
  int tid = blockIdx.x * blockDim.x + threadIdx.x;  // tid = e*4 + h
  if (tid >= EH) return;
  int e = tid >> 2, h = tid & 3;
  int s = src[e], d = dst[e];
  float a = aout[tid] / ssum[d * NHEAD + h];
  aout[tid] = a;  // final attention output
  const float4* frow = (const float4*)(fs + (size_t)s * FS_F + h * SEG_F);
  float* orow = rst + (size_t)d * FS_F + h * SEG_F;
#pragma unroll
  for (int c = 0; c < SEG_F / 4; ++c) {
    float4 v = frow[c];
    atomicAdd(&orow[4 * c + 0], v.x * a);
    atomicAdd(&orow[4 * c + 1], v.y * a);
    atomicAdd(&orow[4 * c + 2], v.z * a);
    atomicAdd(&orow[4 * c + 3], v.w * a);
  }
}

extern "C" void kernel_launch(void* const* d_in, const int* in_sizes, int n_in,
                              void* d_out, int out_size, void* d_ws, size_t ws_size,
                              hipStream_t stream) {
  (void)in_sizes; (void)n_in; (void)out_size; (void)ws_size;
  const float* feat     = (const float*)d_in[0];
  const float* fc       = (const float*)d_in[1];
  const float* attn_l   = (const float*)d_in[2];
  const float* attn_r   = (const float*)d_in[3];
  const float* attn_e   = (const float*)d_in[4];
  const float* edge_emb = (const float*)d_in[5];
  const float* fc_e_w   = (const float*)d_in[6];
  const int*   src      = (const int*)d_in[7];
  const int*   dst      = (const int*)d_in[8];
  const int*   etype    = (const int*)d_in[9];

  float* rst  = (float*)d_out;                      // [N, H, 48]
  float* aout = rst + (size_t)N_NODES * FS_F;       // [E, H]

  // workspace carve-up (~41.6 MB of floats)
  float*        fs      = (float*)d_ws;             // [N, H, 48]
  float*        el      = fs + (size_t)N_NODES * FS_F;
  float*        er      = el + NH;
  unsigned int* mkey    = (unsigned int*)(er + NH);
  float*        ssum    = (float*)(mkey + NH);
  float*        ee_type = ssum + NH;                // [NETYPE, H]

  {
    int total = N_NODES * FS_F;  // also covers NH for ssum/mkey init
    k_init<<<(total + 255) / 256, 256, 0, stream>>>(rst, ssum, mkey);
  }
  k_ee<<<1, 32, 0, stream>>>(edge_emb, fc_e_w, attn_e, ee_type);
  {
    int waves = (N_NODES / 16) * NTYPE;             // 9375 waves
    int blocks = (waves + 3) / 4;                   // 4 waves per 128-thread block
    k_proj<<<blocks, 128, 0, stream>>>(feat, fc, fs);
  }
  k_elr<<<(NH + 255) / 256, 256, 0, stream>>>(fs, attn_l, attn_r, el, er);
  k_score<<<(EH + 255) / 256, 256, 0, stream>>>(src, dst, etype, el, er,
                                                ee_type, aout, mkey);
  k_expsum<<<(EH + 255) / 256, 256, 0, stream>>>(dst, aout, mkey, ssum);
  k_msg<<<(EH + 255) / 256, 256, 0, stream>>>(src, dst, aout, ssum, fs, rst);
}